// PathwayEncoder_25864293057120
// MI455X (gfx1250) — compile-verified
//
#include <hip/hip_runtime.h>

typedef __attribute__((ext_vector_type(2))) float v2f;
typedef __attribute__((ext_vector_type(8))) float v8f;

// ---------------------------------------------------------------------------
// Utility: grid-stride zero
// ---------------------------------------------------------------------------
__global__ void gcn_zero(float* __restrict__ p, long n) {
  long i = (long)blockIdx.x * blockDim.x + threadIdx.x;
  long stride = (long)gridDim.x * blockDim.x;
  for (; i < n; i += stride) p[i] = 0.0f;
}

// ---------------------------------------------------------------------------
// Degree: one atomic per edge (dst side). Self-loop folded into dinv kernel.
// ---------------------------------------------------------------------------
__global__ void gcn_degree(const int* __restrict__ dst, int E,
                           float* __restrict__ deg) {
  int e = blockIdx.x * blockDim.x + threadIdx.x;
  if (e < E) unsafeAtomicAdd(&deg[dst[e]], 1.0f);
}

// dinv[i] = rsqrt(deg[i] + 1)   (in place: deg buffer becomes dinv buffer)
__global__ void gcn_dinv(float* __restrict__ d, int n) {
  int i = blockIdx.x * blockDim.x + threadIdx.x;
  if (i < n) d[i] = rsqrtf(d[i] + 1.0f);
}

// Per-edge normalization weight, shared by BOTH layers:
// w[e] = dinv[src[e]] * dinv[dst[e]]   (computed once, reused twice)
__global__ void gcn_edgew(const int* __restrict__ src,
                          const int* __restrict__ dst,
                          const float* __restrict__ dinv,
                          float* __restrict__ w, int E) {
  int e = blockIdx.x * blockDim.x + threadIdx.x;
  if (e < E) w[e] = dinv[src[e]] * dinv[dst[e]];
}

// ---------------------------------------------------------------------------
// Layer-1 transform: t1[n,16] = X[n,3] @ W1[3,16] via V_WMMA_F32_16X16X4_F32.
// 8 waves per block, one 16-row tile per wave; K=3 padded to 4 with zeros.
// Loads are unconditional (addresses valid for all lanes) and the K-pad is
// applied with a branch-free select, so EXEC is all-1s at the WMMA.
//   A (16x4 f32): lanes 0-15 -> {K0,K1}, lanes 16-31 -> {K2,K3}
//   B (4x16 f32): lanes 0-15 -> rows {K0,K1}, lanes 16-31 -> rows {K2,K3}
//   C/D (16x16 f32): vgpr v, lanes 0-15: M=v; lanes 16-31: M=v+8
// ---------------------------------------------------------------------------
__global__ void gcn_xw1_wmma(const float* __restrict__ x,
                             const float* __restrict__ W1,
                             float* __restrict__ t1, int ntiles) {
  const int wave = threadIdx.x >> 5;                 // 0..7
  const int tile = blockIdx.x * (blockDim.x >> 5) + wave;
  if (tile >= ntiles) return;                        // wave-uniform guard
  const int lane = threadIdx.x & 31;
  const int half = lane >> 4;                        // 0: K={0,1}  1: K={2,3}
  const int l    = lane & 15;
  const int row  = tile * 16 + l;

  // Unconditional loads; mask the K=3 / row-3 zero pad with selects.
  float ax = x[row * 3 + 2 * half];                  // K0 or K2
  float ay = x[row * 3 + 1];                         // K1 (valid addr for all)
  float bx = W1[(2 * half) * 16 + l];                // W1 row 0 or row 2
  float by = W1[1 * 16 + l];                         // W1 row 1
  v2f a, b;
  a.x = ax;
  a.y = (half == 0) ? ay : 0.0f;
  b.x = bx;
  b.y = (half == 0) ? by : 0.0f;

  v8f c = {};
  c = __builtin_amdgcn_wmma_f32_16x16x4_f32(false, a, false, b,
                                            (short)0, c, false, false);
#pragma unroll
  for (int i = 0; i < 8; ++i) {
    int m = i + half * 8;
    t1[(tile * 16 + m) * 16 + l] = c[i];
  }
}

// ---------------------------------------------------------------------------
// Layer-2 transform: t2[n,32] = y1[n,16] @ W2[16,32].
// K=16 -> 4 chunks of K=4; N=32 -> two 16-wide accumulators. 8 WMMAs/tile.
// ---------------------------------------------------------------------------
__global__ void gcn_hw2_wmma(const float* __restrict__ y1,
                             const float* __restrict__ W2,
                             float* __restrict__ t2, int ntiles) {
  const int wave = threadIdx.x >> 5;
  const int tile = blockIdx.x * (blockDim.x >> 5) + wave;
  if (tile >= ntiles) return;                        // wave-uniform guard
  const int lane = threadIdx.x & 31;
  const int half = lane >> 4;
  const int l    = lane & 15;
  const int row  = tile * 16 + l;

  v8f c0 = {}, c1 = {};
#pragma unroll
  for (int kk = 0; kk < 4; ++kk) {
    const int k0 = kk * 4 + 2 * half;                // this lane's first K
    v2f a, b0, b1;
    a.x = y1[row * 16 + k0 + 0];
    a.y = y1[row * 16 + k0 + 1];
    b0.x = W2[(k0 + 0) * 32 + 0 * 16 + l];
    b0.y = W2[(k0 + 1) * 32 + 0 * 16 + l];
    b1.x = W2[(k0 + 0) * 32 + 1 * 16 + l];
    b1.y = W2[(k0 + 1) * 32 + 1 * 16 + l];
    c0 = __builtin_amdgcn_wmma_f32_16x16x4_f32(false, a, false, b0,
                                               (short)0, c0, false, false);
    c1 = __builtin_amdgcn_wmma_f32_16x16x4_f32(false, a, false, b1,
                                               (short)0, c1, false, false);
  }
#pragma unroll
  for (int i = 0; i < 8; ++i) {
    int m = i + half * 8;
    int r = tile * 16 + m;
    t2[r * 32 + l]      = c0[i];
    t2[r * 32 + 16 + l] = c1[i];
  }
}

// ---------------------------------------------------------------------------
// Edge scatter, layer 1 (F=16): one thread per (edge, feature).
// 16 consecutive lanes cover one edge -> coalesced 64B gather of t1[src,:]
// and contiguous atomics into agg1[dst,:]. w[e] is a single broadcast load.
// All tables are L2-resident (192 MB L2), so random traffic stays on-chip.
// ---------------------------------------------------------------------------
__global__ void gcn_scatter16(const int* __restrict__ src,
                              const int* __restrict__ dst,
                              const float* __restrict__ w,
                              const float* __restrict__ t1,
                              float* __restrict__ agg1, long E) {
  long t = (long)blockIdx.x * blockDim.x + threadIdx.x;
  if (t >= E * 16) return;
  int e = (int)(t >> 4);
  int f = (int)(t & 15);
  int s = src[e], d = dst[e];
  unsafeAtomicAdd(&agg1[(long)d * 16 + f], t1[(long)s * 16 + f] * w[e]);
}

// Self-loop + bias + ReLU, in place (agg1 -> y1)
__global__ void gcn_relu1(float* __restrict__ agg1,
                          const float* __restrict__ t1,
                          const float* __restrict__ dinv,
                          const float* __restrict__ b1, long n16) {
  long t = (long)blockIdx.x * blockDim.x + threadIdx.x;
  if (t >= n16) return;
  int i = (int)(t >> 4);
  int f = (int)(t & 15);
  float di = dinv[i];
  float v = agg1[t] + t1[t] * di * di + b1[f];
  agg1[t] = v > 0.0f ? v : 0.0f;
}

// Edge scatter, layer 2 (F=32)
__global__ void gcn_scatter32(const int* __restrict__ src,
                              const int* __restrict__ dst,
                              const float* __restrict__ w,
                              const float* __restrict__ t2,
                              float* __restrict__ agg2, long E) {
  long t = (long)blockIdx.x * blockDim.x + threadIdx.x;
  if (t >= E * 32) return;
  int e = (int)(t >> 5);
  int f = (int)(t & 31);
  int s = src[e], d = dst[e];
  unsafeAtomicAdd(&agg2[(long)d * 32 + f], t2[(long)s * 32 + f] * w[e]);
}

// ---------------------------------------------------------------------------
// Finalize layer 2 (self-loop + bias) and mean-pool over nodes into out[32].
// Each thread owns one feature lane and a strided slice of nodes (a warp
// covers one full 32-float row per iteration -> coalesced); one atomic per
// thread at the end. b2 is added exactly once (by the chunk-0 thread).
// ---------------------------------------------------------------------------
__global__ void gcn_mean(const float* __restrict__ agg2,
                         const float* __restrict__ t2,
                         const float* __restrict__ dinv,
                         const float* __restrict__ b2,
                         float* __restrict__ out, int n) {
  int t = blockIdx.x * blockDim.x + threadIdx.x;
  int f = t & 31;
  int chunk = t >> 5;
  int nchunks = (gridDim.x * blockDim.x) >> 5;
  float acc = 0.0f;
  for (int i = chunk; i < n; i += nchunks) {
    float di = dinv[i];
    acc += agg2[(long)i * 32 + f] + t2[(long)i * 32 + f] * di * di;
  }
  float res = acc * (1.0f / (float)n);
  if (chunk == 0) res += b2[f];
  unsafeAtomicAdd(&out[f], res);
}

// ---------------------------------------------------------------------------
// Launch sequence (all on `stream`, deterministic, graph-capture safe)
// ---------------------------------------------------------------------------
extern "C" void kernel_launch(void* const* d_in, const int* in_sizes, int n_in,
                              void* d_out, int out_size, void* d_ws, size_t ws_size,
                              hipStream_t stream) {
  const float* x  = (const float*)d_in[0];   // [n,3]
  const int*   ei = (const int*)d_in[1];     // [2,E] flat
  const float* W1 = (const float*)d_in[2];   // [3,16]
  const float* b1 = (const float*)d_in[3];   // [16]
  const float* W2 = (const float*)d_in[4];   // [16,32]
  const float* b2 = (const float*)d_in[5];   // [32]
  float* out = (float*)d_out;                // [32]

  const int  n = in_sizes[0] / 3;            // 100000
  const long E = in_sizes[1] / 2;            // 2400000
  const int* src = ei;
  const int* dst = ei + E;

  // Workspace layout (floats): dinv | t1 | agg1 | t2 | agg2 | w
  float* ws   = (float*)d_ws;
  float* dinv = ws;                          // n       (deg -> dinv in place)
  float* t1   = dinv + n;                    // 16n
  float* agg1 = t1 + (long)16 * n;           // 16n (becomes y1 after ReLU)
  float* t2   = agg1 + (long)16 * n;         // 32n
  float* agg2 = t2 + (long)32 * n;           // 32n
  float* w    = agg2 + (long)32 * n;         // E   (per-edge norm weight)

  const int B = 256;
  const int ntiles = n / 16;                 // n % 16 == 0 for this problem
  const int wgrid  = (ntiles + 7) / 8;       // 8 waves (tiles) per block

  // 1) zero accumulators + output
  gcn_zero<<<512, B, 0, stream>>>(dinv, (long)n);
  gcn_zero<<<2048, B, 0, stream>>>(agg1, (long)16 * n);
  gcn_zero<<<4096, B, 0, stream>>>(agg2, (long)32 * n);
  gcn_zero<<<1, 32, 0, stream>>>(out, 32);

  // 2) degrees -> dinv -> per-edge weights
  gcn_degree<<<(int)((E + B - 1) / B), B, 0, stream>>>(dst, (int)E, dinv);
  gcn_dinv<<<(n + B - 1) / B, B, 0, stream>>>(dinv, n);
  gcn_edgew<<<(int)((E + B - 1) / B), B, 0, stream>>>(src, dst, dinv, w, (int)E);

  // 3) t1 = X @ W1  (WMMA)
  gcn_xw1_wmma<<<wgrid, B, 0, stream>>>(x, W1, t1, ntiles);

  // 4) layer-1 scatter + self-loop/bias/ReLU
  long t1th = E * 16;
  gcn_scatter16<<<(int)((t1th + B - 1) / B), B, 0, stream>>>(src, dst, w, t1, agg1, E);
  gcn_relu1<<<(int)(((long)16 * n + B - 1) / B), B, 0, stream>>>(agg1, t1, dinv, b1, (long)16 * n);

  // 5) t2 = y1 @ W2  (WMMA, 8 wmma per tile)
  gcn_hw2_wmma<<<wgrid, B, 0, stream>>>(agg1, W2, t2, ntiles);

  // 6) layer-2 scatter
  long t2th = E * 32;
  gcn_scatter32<<<(int)((t2th + B - 1) / B), B, 0, stream>>>(src, dst, w, t2, agg2, E);

  // 7) finalize + mean pool
  gcn_mean<<<512, B, 0, stream>>>(agg2, t2, dinv, b2, out, n);
}